// G2_53712861003961
// MI455X (gfx1250) — compile-verified
//
#include <hip/hip_runtime.h>
#include <hip/hip_bf16.h>

typedef __attribute__((ext_vector_type(2))) float v2f;
typedef __attribute__((ext_vector_type(8))) float v8f;

// LDS pair-interleaved weight layout:
//   element W[r][c]  ->  lds[(r>>1)*LSTRIDE + c*2 + (r&1)]
// so the WMMA B fragment {W[r][c], W[r+1][c]} (r even) is one aligned b64.
// LSTRIDE = 160 dwords: 160 mod 64 = 32, so the two half-wave groups
// (k-pair rows differing by 1) hit disjoint LDS bank sets -> conflict-free.
#define LSTRIDE 160
#define LDS_W_FLOATS (32 * LSTRIDE)

__device__ __forceinline__ void stage_weight_lds(const float* __restrict__ W,
                                                 float* __restrict__ lds,
                                                 int tid) {
    const float4* w4 = (const float4*)W;
    for (int i = tid; i < 1024; i += 256) {   // 4096 floats per matrix
        float4 v = w4[i];
        int f0 = i * 4;
        int r = f0 >> 6;
        int c0 = f0 & 63;
        int base = (r >> 1) * LSTRIDE + (r & 1);
        lds[base + (c0 + 0) * 2] = v.x;
        lds[base + (c0 + 1) * 2] = v.y;
        lds[base + (c0 + 2) * 2] = v.z;
        lds[base + (c0 + 3) * 2] = v.w;
    }
}

// ---------------------------------------------------------------------------
// Kernel 0: repack Qw [4,128,16] into U[64][64] (src half) and V[64][64]
// (dst half), flattening head/out dims: out col k = h*16 + d.
// ---------------------------------------------------------------------------
__global__ __launch_bounds__(256) void gsage_repack(const float* __restrict__ Qw,
                                                    float* __restrict__ U,
                                                    float* __restrict__ V) {
    int t = blockIdx.x * 256 + threadIdx.x;
    if (t >= 2 * 4096) return;
    int which = t >> 12;       // 0 -> U, 1 -> V
    int idx   = t & 4095;
    int f = idx >> 6;          // input-feature row 0..63
    int k = idx & 63;          // flattened out col
    int h = k >> 4, d = k & 15;
    int frow = which ? (64 + f) : f;
    float val = Qw[h * 2048 + frow * 16 + d];
    if (which) V[idx] = val; else U[idx] = val;
}

// ---------------------------------------------------------------------------
// Kernel 1: per-edge scatter: aggsum[dst] += X[src]; degree counts.
// 64 threads per edge (one per feature column).
// ---------------------------------------------------------------------------
__global__ __launch_bounds__(256) void gsage_edge_agg(const int* __restrict__ src,
                                                      const int* __restrict__ dst,
                                                      const float* __restrict__ X,
                                                      float* __restrict__ aggsum,
                                                      float* __restrict__ degd,
                                                      float* __restrict__ degs,
                                                      long long Ethreads) {
    long long t = (long long)blockIdx.x * 256 + threadIdx.x;
    if (t >= Ethreads) return;
    int e = (int)(t >> 6);
    int c = (int)(t & 63);
    int s = src[e], d = dst[e];
    atomicAdd(&aggsum[(long long)d * 64 + c], X[(long long)s * 64 + c]);
    if (c == 0) {
        atomicAdd(&degd[d], 1.0f);
        atomicAdd(&degs[s], 1.0f);
    }
}

// ---------------------------------------------------------------------------
// Kernel 2: X2 = relu(X @ W_self + (aggsum/deg) @ W_neigh + b)  via
// V_WMMA_F32_16X16X4_F32. One wave32 per 16-row tile; weights staged in LDS.
// A layout (16x4 f32): lane L holds row (L&15), K pair 2*(L>>4).
// B layout (4x16):    VGPR j, lane L -> row j + 2*(L>>4), col (L&15).
// C layout (16x16):   VGPR j, lane L -> row j + 8*(L>>4), col (L&15).
// ---------------------------------------------------------------------------
__global__ __launch_bounds__(256) void gsage_conv_gemm(const float* __restrict__ X,
                                                       const float* __restrict__ aggsum,
                                                       const float* __restrict__ degd,
                                                       const float* __restrict__ Wself,
                                                       const float* __restrict__ Wneigh,
                                                       const float* __restrict__ bconv,
                                                       float* __restrict__ X2,
                                                       int N) {
    __shared__ float w1[LDS_W_FLOATS];
    __shared__ float w2[LDS_W_FLOATS];
    int tid = threadIdx.x;
    stage_weight_lds(Wself, w1, tid);
    stage_weight_lds(Wneigh, w2, tid);
    __syncthreads();

    int wave = tid >> 5;
    int lane = tid & 31;
    int tile = blockIdx.x * 8 + wave;
    int row0 = tile * 16;
    if (row0 >= N) return;                 // wave-uniform: EXEC stays all-1s

    int mr    = lane & 15;                 // A row within tile / C col within nt
    int khalf = (lane >> 4) * 2;           // K sub-pair select (0 or 2)
    int row   = row0 + mr;
    float invd = 1.0f / fmaxf(degd[row], 1.0f);
    const float* xrow = X + (long long)row * 64;
    const float* arow = aggsum + (long long)row * 64;

    v8f acc[4];
#pragma unroll
    for (int nt = 0; nt < 4; ++nt) acc[nt] = (v8f)0.0f;

    for (int kk = 0; kk < 16; ++kk) {
        int kb = kk * 4 + khalf;                    // even
        v2f ax = *(const v2f*)(xrow + kb);          // global_load_b64
        v2f aa = *(const v2f*)(arow + kb);
        aa *= invd;
        int bbase = (kk * 2 + (lane >> 4)) * LSTRIDE;   // (kb>>1) row
#pragma unroll
        for (int nt = 0; nt < 4; ++nt) {
            int c = nt * 16 + mr;
            v2f b1 = *(const v2f*)(w1 + bbase + c * 2);  // ds_load_b64
            v2f b2 = *(const v2f*)(w2 + bbase + c * 2);
            acc[nt] = __builtin_amdgcn_wmma_f32_16x16x4_f32(false, ax, false, b1,
                                                            (short)0, acc[nt], false, false);
            acc[nt] = __builtin_amdgcn_wmma_f32_16x16x4_f32(false, aa, false, b2,
                                                            (short)0, acc[nt], false, false);
        }
    }
    // epilogue: +bias, relu, store
    int mbase = row0 + (lane >> 4) * 8;
#pragma unroll
    for (int nt = 0; nt < 4; ++nt) {
        int c = nt * 16 + mr;
        float bias = bconv[c];
#pragma unroll
        for (int j = 0; j < 8; ++j) {
            float v = fmaxf(acc[nt][j] + bias, 0.0f);
            X2[(long long)(mbase + j) * 64 + c] = v;
        }
    }
}

// ---------------------------------------------------------------------------
// Kernel 3: Aout = X2 @ U + Qb ; Y = X2 @ V   (same WMMA tiling)
// ---------------------------------------------------------------------------
__global__ __launch_bounds__(256) void gsage_quad_gemm(const float* __restrict__ X2,
                                                       const float* __restrict__ U,
                                                       const float* __restrict__ V,
                                                       const float* __restrict__ Qb,
                                                       float* __restrict__ Aout,
                                                       float* __restrict__ Yout,
                                                       int N) {
    __shared__ float uL[LDS_W_FLOATS];
    __shared__ float vL[LDS_W_FLOATS];
    int tid = threadIdx.x;
    stage_weight_lds(U, uL, tid);
    stage_weight_lds(V, vL, tid);
    __syncthreads();

    int wave = tid >> 5;
    int lane = tid & 31;
    int tile = blockIdx.x * 8 + wave;
    int row0 = tile * 16;
    if (row0 >= N) return;

    int mr    = lane & 15;
    int khalf = (lane >> 4) * 2;
    int row   = row0 + mr;
    const float* xrow = X2 + (long long)row * 64;

    v8f accP[4], accY[4];
#pragma unroll
    for (int nt = 0; nt < 4; ++nt) { accP[nt] = (v8f)0.0f; accY[nt] = (v8f)0.0f; }

    for (int kk = 0; kk < 16; ++kk) {
        int kb = kk * 4 + khalf;
        v2f ax = *(const v2f*)(xrow + kb);
        int bbase = (kk * 2 + (lane >> 4)) * LSTRIDE;
#pragma unroll
        for (int nt = 0; nt < 4; ++nt) {
            int c = nt * 16 + mr;
            v2f bu = *(const v2f*)(uL + bbase + c * 2);
            v2f bv = *(const v2f*)(vL + bbase + c * 2);
            accP[nt] = __builtin_amdgcn_wmma_f32_16x16x4_f32(false, ax, false, bu,
                                                             (short)0, accP[nt], false, false);
            accY[nt] = __builtin_amdgcn_wmma_f32_16x16x4_f32(false, ax, false, bv,
                                                             (short)0, accY[nt], false, false);
        }
    }
    int mbase = row0 + (lane >> 4) * 8;
#pragma unroll
    for (int nt = 0; nt < 4; ++nt) {
        int c = nt * 16 + mr;
        float qb = Qb[c];
#pragma unroll
        for (int j = 0; j < 8; ++j) {
            long long o = (long long)(mbase + j) * 64 + c;
            Aout[o] = accP[nt][j] + qb;   // alpha + bias, per node
            Yout[o] = accY[nt][j];        // beta basis, per node
        }
    }
}

// ---------------------------------------------------------------------------
// Kernel 4: per-edge accumulate S1[src] += Y[dst], S2[src] += Y[dst]^2
// ---------------------------------------------------------------------------
__global__ __launch_bounds__(256) void gsage_edge_quad(const int* __restrict__ src,
                                                       const int* __restrict__ dst,
                                                       const float* __restrict__ Y,
                                                       float* __restrict__ S1,
                                                       float* __restrict__ S2,
                                                       long long Ethreads) {
    long long t = (long long)blockIdx.x * 256 + threadIdx.x;
    if (t >= Ethreads) return;
    int e = (int)(t >> 6);
    int c = (int)(t & 63);
    int s = src[e], d = dst[e];
    float y = Y[(long long)d * 64 + c];
    atomicAdd(&S1[(long long)s * 64 + c], y);
    atomicAdd(&S2[(long long)s * 64 + c], y * y);
}

// ---------------------------------------------------------------------------
// Kernel 5: gg = tanh((deg*a^2 + 2*a*S1 + S2) / max(deg,1))
// ---------------------------------------------------------------------------
__global__ __launch_bounds__(256) void gsage_final(const float* __restrict__ Aout,
                                                   const float* __restrict__ S1,
                                                   const float* __restrict__ S2,
                                                   const float* __restrict__ degs,
                                                   float* __restrict__ out,
                                                   long long total) {
    long long t = (long long)blockIdx.x * 256 + threadIdx.x;
    if (t >= total) return;
    int n = (int)(t >> 6);
    float deg = degs[n];
    float a = Aout[t];
    float m = (deg * a * a + 2.0f * a * S1[t] + S2[t]) / fmaxf(deg, 1.0f);
    out[t] = tanhf(m);
}

// ---------------------------------------------------------------------------
extern "C" void kernel_launch(void* const* d_in, const int* in_sizes, int n_in,
                              void* d_out, int out_size, void* d_ws, size_t ws_size,
                              hipStream_t stream) {
    const float* X      = (const float*)d_in[0];
    const int*   ei     = (const int*)d_in[1];
    const float* Wself  = (const float*)d_in[2];
    const float* Wneigh = (const float*)d_in[3];
    const float* bconv  = (const float*)d_in[4];
    const float* Qw     = (const float*)d_in[5];
    const float* Qb     = (const float*)d_in[6];

    const int N = in_sizes[0] / 64;     // nhid = 64
    const int E = in_sizes[1] / 2;
    const int* src = ei;
    const int* dst = ei + E;

    // Workspace layout (floats). Zeroed region first, contiguous.
    float* ws = (float*)d_ws;
    size_t off = 0;
    float* aggsum = ws + off; off += (size_t)N * 64;   // reused later as Aout
    float* S1     = ws + off; off += (size_t)N * 64;
    float* S2     = ws + off; off += (size_t)N * 64;
    float* degd   = ws + off; off += (size_t)N;
    float* degs   = ws + off; off += (size_t)N;
    float* X2     = ws + off; off += (size_t)N * 64;
    float* Yout   = ws + off; off += (size_t)N * 64;
    float* U      = ws + off; off += 4096;
    float* V      = ws + off; off += 4096;
    float* Aout   = aggsum;  // aggsum dead after gsage_conv_gemm

    // zero aggsum, S1, S2, degd, degs  (194*N floats, contiguous at ws start)
    hipMemsetAsync(d_ws, 0, (size_t)194 * N * sizeof(float), stream);

    gsage_repack<<<32, 256, 0, stream>>>(Qw, U, V);

    long long Ethr = (long long)E * 64;
    int eblocks = (int)((Ethr + 255) / 256);
    gsage_edge_agg<<<eblocks, 256, 0, stream>>>(src, dst, X, aggsum, degd, degs, Ethr);

    int tiles = (N + 15) / 16;          // N = 100000 is a multiple of 16
    int gblocks = (tiles + 7) / 8;
    gsage_conv_gemm<<<gblocks, 256, 0, stream>>>(X, aggsum, degd, Wself, Wneigh,
                                                 bconv, X2, N);
    gsage_quad_gemm<<<gblocks, 256, 0, stream>>>(X2, U, V, Qb, Aout, Yout, N);

    gsage_edge_quad<<<eblocks, 256, 0, stream>>>(src, dst, Yout, S1, S2, Ethr);

    long long total = (long long)N * 64;
    int fblocks = (int)((total + 255) / 256);
    gsage_final<<<fblocks, 256, 0, stream>>>(Aout, S1, S2, degs, (float*)d_out, total);
}